// OTLoss_38766374813948
// MI455X (gfx1250) — compile-verified
//
#include <hip/hip_runtime.h>
#include <hip/hip_bf16.h>

typedef __attribute__((ext_vector_type(2))) float v2f;
typedef __attribute__((ext_vector_type(8))) float v8f;

#define N_PTS 4096
#define DIMS 1024
#define EPS_F 0.01f
#define INV_EPS 100.0f
#define INV_ENTREG 10.0f
#define NEG_LOG_N (-8.317766166719343f) /* -log(4096) */
#define SINK_ITERS 30

// ---------------------------------------------------------------- row norms
__global__ __launch_bounds__(256) void rownorm_kernel(const float* __restrict__ X,
                                                      float* __restrict__ out) {
  const int row = blockIdx.x;
  const float* p = X + (size_t)row * DIMS;
  float s = 0.f;
  for (int c = threadIdx.x; c < DIMS; c += 256) {
    float v = p[c];
    s += v * v;
  }
  __shared__ float red[8];
#pragma unroll
  for (int off = 16; off; off >>= 1) s += __shfl_xor(s, off, 32);
  if ((threadIdx.x & 31) == 0) red[threadIdx.x >> 5] = s;
  __syncthreads();
  if (threadIdx.x == 0) {
    float v = 0.f;
#pragma unroll
    for (int w = 0; w < 8; ++w) v += red[w];
    out[row] = v;
  }
}

__global__ void init_fg_kernel(float* __restrict__ f, float* __restrict__ g) {
  int i = blockIdx.x * 256 + threadIdx.x;
  if (i < N_PTS) {
    f[i] = 0.f;
    g[i] = 0.f;
  }
}

// ------------------------------------------------- cost matrix via f32 WMMA
// C[i][j] = x2[i] + y2[j] - 2 * dot(Xs_i, Xt_j)
#define TILE 128
#define KBLK 32
#define KP (KBLK / 2)     // k-pair rows
#define PSTRIDE (TILE + 16)  // v2f units; dword stride 288 % 64 == 32 -> halves on disjoint banks

__global__ __launch_bounds__(256) void gemm_cost_kernel(
    const float* __restrict__ Xs, const float* __restrict__ Xt,
    const float* __restrict__ x2, const float* __restrict__ y2,
    float* __restrict__ C) {
  // K-paired layout: element (k, m) lives in sA2[k>>1][m][k&1]
  __shared__ v2f sA2[KP][PSTRIDE];
  __shared__ v2f sB2[KP][PSTRIDE];
  const int tid = threadIdx.x;
  const int lane = tid & 31;
  const int wv = tid >> 5;
  const int waveM = wv & 3;   // 4 row-groups of 32
  const int waveN = wv >> 2;  // 2 col-groups of 64
  const int row0 = blockIdx.y * TILE;
  const int col0 = blockIdx.x * TILE;
  const int lhalf = lane >> 4;  // 0/1: selects k-pair (A/B) and M+8 (C/D)
  const int l15 = lane & 15;

  v8f acc[2][4];
#pragma unroll
  for (int rt = 0; rt < 2; ++rt)
#pragma unroll
    for (int bt = 0; bt < 4; ++bt)
      acc[rt][bt] = (v8f){0.f, 0.f, 0.f, 0.f, 0.f, 0.f, 0.f, 0.f};

  for (int k0 = 0; k0 < DIMS; k0 += KBLK) {
    __syncthreads();
#pragma unroll
    for (int i = 0; i < 4; ++i) {
      int idx = tid + 256 * i;
      int r = idx >> 3;
      int c4 = (idx & 7) << 2;        // k column group (multiple of 4)
      int p0 = c4 >> 1;               // k-pair row
      float4 a = *(const float4*)(Xs + (size_t)(row0 + r) * DIMS + k0 + c4);
      sA2[p0 + 0][r] = (v2f){a.x, a.y};
      sA2[p0 + 1][r] = (v2f){a.z, a.w};
      float4 b = *(const float4*)(Xt + (size_t)(col0 + r) * DIMS + k0 + c4);
      sB2[p0 + 0][r] = (v2f){b.x, b.y};
      sB2[p0 + 1][r] = (v2f){b.z, b.w};
    }
    __syncthreads();
    for (int kk2 = 0; kk2 < KP; kk2 += 2) {   // each step covers K=4
      const int kp = kk2 + lhalf;  // lane-half picks its k-pair
      v2f af[2], bf[4];
#pragma unroll
      for (int rt = 0; rt < 2; ++rt)
        af[rt] = sA2[kp][waveM * 32 + rt * 16 + l15];   // single ds_load_b64
#pragma unroll
      for (int bt = 0; bt < 4; ++bt)
        bf[bt] = sB2[kp][waveN * 64 + bt * 16 + l15];
#pragma unroll
      for (int rt = 0; rt < 2; ++rt)
#pragma unroll
        for (int bt = 0; bt < 4; ++bt)
          acc[rt][bt] = __builtin_amdgcn_wmma_f32_16x16x4_f32(
              false, af[rt], false, bf[bt], (short)0, acc[rt][bt], false, false);
    }
  }
  // epilogue: C = x2 + y2 - 2*dot
#pragma unroll
  for (int rt = 0; rt < 2; ++rt) {
#pragma unroll
    for (int bt = 0; bt < 4; ++bt) {
      int n = col0 + waveN * 64 + bt * 16 + l15;
      float yv = y2[n];
#pragma unroll
      for (int v = 0; v < 8; ++v) {
        int m = row0 + waveM * 32 + rt * 16 + (lhalf << 3) + v;
        C[(size_t)m * N_PTS + n] = x2[m] + yv - 2.0f * acc[rt][bt][v];
      }
    }
  }
}

// ---------------------------------------------- online logsumexp machinery
__device__ __forceinline__ void lse_merge(float& m, float& s, float zm, float zs) {
  float nm = fmaxf(m, zm);
  s = s * __expf(m - nm) + zs * __expf(zm - nm);
  m = nm;
}

// g update, stage A: 4 columns per thread (b128 loads), 256-row chunks.
__global__ __launch_bounds__(256) void col_lse_partial(const float* __restrict__ C,
                                                       const float* __restrict__ f,
                                                       float* __restrict__ pm,
                                                       float* __restrict__ ps) {
  const int j4 = (blockIdx.x * 256 + threadIdx.x) << 2;  // grid.x = 4
  const int chunk = blockIdx.y;                          // 0..15
  const int i0 = chunk * 256;
  float m[4], s[4];
#pragma unroll
  for (int q = 0; q < 4; ++q) {
    m[q] = -INFINITY;
    s[q] = 0.f;
  }
  for (int i = i0; i < i0 + 256; ++i) {
    float fi = f[i];
    float4 c = *(const float4*)(C + (size_t)i * N_PTS + j4);
    lse_merge(m[0], s[0], (fi - c.x) * INV_EPS + NEG_LOG_N, 1.0f);
    lse_merge(m[1], s[1], (fi - c.y) * INV_EPS + NEG_LOG_N, 1.0f);
    lse_merge(m[2], s[2], (fi - c.z) * INV_EPS + NEG_LOG_N, 1.0f);
    lse_merge(m[3], s[3], (fi - c.w) * INV_EPS + NEG_LOG_N, 1.0f);
  }
  *(float4*)(pm + chunk * N_PTS + j4) = make_float4(m[0], m[1], m[2], m[3]);
  *(float4*)(ps + chunk * N_PTS + j4) = make_float4(s[0], s[1], s[2], s[3]);
}

__global__ __launch_bounds__(256) void col_lse_combine(const float* __restrict__ pm,
                                                       const float* __restrict__ ps,
                                                       float* __restrict__ g) {
  const int j = blockIdx.x * 256 + threadIdx.x;
  float m = -INFINITY, s = 0.f;
  for (int c = 0; c < 16; ++c) lse_merge(m, s, pm[c * N_PTS + j], ps[c * N_PTS + j]);
  g[j] = -EPS_F * (m + __logf(s));
}

// f update, stage A: one wave per (row, 2048-col chunk); b128 per lane.
__global__ __launch_bounds__(256) void row_lse_partial(const float* __restrict__ C,
                                                       const float* __restrict__ g,
                                                       float* __restrict__ pm,
                                                       float* __restrict__ ps) {
  const int wv = threadIdx.x >> 5, lane = threadIdx.x & 31;
  const int i = blockIdx.x * 8 + wv;
  const int chunk = blockIdx.y;  // 0..1
  const int j0 = chunk * 2048;
  const float* Crow = C + (size_t)i * N_PTS;
  float m = -INFINITY, s = 0.f;
  for (int jj = 0; jj < 2048; jj += 128) {
    int j = j0 + jj + (lane << 2);
    float4 c = *(const float4*)(Crow + j);
    float4 gv = *(const float4*)(g + j);
    lse_merge(m, s, (gv.x - c.x) * INV_EPS + NEG_LOG_N, 1.0f);
    lse_merge(m, s, (gv.y - c.y) * INV_EPS + NEG_LOG_N, 1.0f);
    lse_merge(m, s, (gv.z - c.z) * INV_EPS + NEG_LOG_N, 1.0f);
    lse_merge(m, s, (gv.w - c.w) * INV_EPS + NEG_LOG_N, 1.0f);
  }
#pragma unroll
  for (int off = 16; off; off >>= 1) {
    float om = __shfl_xor(m, off, 32);
    float os = __shfl_xor(s, off, 32);
    lse_merge(m, s, om, os);
  }
  if (lane == 0) {
    pm[chunk * N_PTS + i] = m;
    ps[chunk * N_PTS + i] = s;
  }
}

__global__ __launch_bounds__(256) void row_lse_combine(const float* __restrict__ pm,
                                                       const float* __restrict__ ps,
                                                       float* __restrict__ f) {
  const int i = blockIdx.x * 256 + threadIdx.x;
  float m = -INFINITY, s = 0.f;
  for (int c = 0; c < 2; ++c) lse_merge(m, s, pm[c * N_PTS + i], ps[c * N_PTS + i]);
  f[i] = -EPS_F * (m + __logf(s));
}

// --------------------------------------------- final transport plan, in place
__global__ __launch_bounds__(256) void plan_kernel(float* __restrict__ C,
                                                   const float* __restrict__ f,
                                                   const float* __restrict__ g) {
  const int idx4 = (blockIdx.x * 256 + threadIdx.x) << 2;
  const int i = idx4 >> 12;
  const int j = idx4 & (N_PTS - 1);
  const float fi = f[i];
  float4 gv = *(const float4*)(g + j);
  float4 c = *(float4*)(C + idx4);
  c.x = __expf((fi + gv.x - c.x) * INV_ENTREG);
  c.y = __expf((fi + gv.y - c.y) * INV_ENTREG);
  c.z = __expf((fi + gv.z - c.z) * INV_ENTREG);
  c.w = __expf((fi + gv.w - c.w) * INV_ENTREG);
  *(float4*)(C + idx4) = c;
}

extern "C" void kernel_launch(void* const* d_in, const int* in_sizes, int n_in,
                              void* d_out, int out_size, void* d_ws, size_t ws_size,
                              hipStream_t stream) {
  const float* Xs = (const float*)d_in[0];
  const float* Xt = (const float*)d_in[1];
  float* C = (float*)d_out;  // 4096*4096 f32: holds cost during iters, P at end
  float* ws = (float*)d_ws;
  float* x2 = ws;             // 4096
  float* y2 = ws + 4096;      // 4096
  float* f = ws + 8192;       // 4096
  float* g = ws + 12288;      // 4096
  float* pm = ws + 16384;     // 16*4096 partial maxes
  float* ps = ws + 81920;     // 16*4096 partial sums

  rownorm_kernel<<<N_PTS, 256, 0, stream>>>(Xs, x2);
  rownorm_kernel<<<N_PTS, 256, 0, stream>>>(Xt, y2);
  init_fg_kernel<<<16, 256, 0, stream>>>(f, g);

  dim3 gemm_grid(N_PTS / TILE, N_PTS / TILE);
  gemm_cost_kernel<<<gemm_grid, 256, 0, stream>>>(Xs, Xt, x2, y2, C);

  for (int it = 0; it < SINK_ITERS; ++it) {
    col_lse_partial<<<dim3(4, 16), 256, 0, stream>>>(C, f, pm, ps);
    col_lse_combine<<<16, 256, 0, stream>>>(pm, ps, g);
    row_lse_partial<<<dim3(512, 2), 256, 0, stream>>>(C, g, pm, ps);
    row_lse_combine<<<16, 256, 0, stream>>>(pm, ps, f);
  }

  plan_kernel<<<(N_PTS * N_PTS) / 1024, 256, 0, stream>>>(C, f, g);
}